// LogitTransform_29703993819785
// MI455X (gfx1250) — compile-verified
//
#include <hip/hip_runtime.h>
#include <hip/hip_bf16.h>
#include <math.h>

// Problem constants (from reference)
#define BB 2
#define SS 512
#define EE 128
#define VV 50000

typedef float v2f __attribute__((ext_vector_type(2)));
typedef float v4f __attribute__((ext_vector_type(4)));
typedef float v8f __attribute__((ext_vector_type(8)));

// ---------------------------------------------------------------------------
// Kernel 0: streaming zero-fill of d_out (204.8 MB -> bandwidth floor ~9us).
// Non-temporal so we don't pollute L2 (array > 192MB L2 anyway).
// ---------------------------------------------------------------------------
__global__ void zero_out_kernel(float* __restrict__ out, size_t n4) {
    v4f* o4 = (v4f*)out;
    size_t i = (size_t)blockIdx.x * blockDim.x + threadIdx.x;
    size_t stride = (size_t)gridDim.x * blockDim.x;
    const v4f z = {0.f, 0.f, 0.f, 0.f};
    for (; i < n4; i += stride) {
        __builtin_nontemporal_store(z, &o4[i]);
    }
}

// ---------------------------------------------------------------------------
// Kernel 1: emb = H @ W_emb^T + b_emb ; ptr = H @ W_ptr^T + b_ptr
// M = B*S = 1024, N = K = 128.  One wave per 16x16 C tile, fp32 WMMA.
// A[m][k] per lane: m = l&15, k = k0 + 2*(l>>4) + j  (v2f)
// B[k][n] = W[n][k] per lane: n = l&15  -> contiguous float2 from W row.
// blockIdx.z selects (W_emb,b_emb)->emb or (W_ptr,b_ptr)->ptr.
// ---------------------------------------------------------------------------
__global__ __launch_bounds__(32)
void gemm_linear_wmma(const float* __restrict__ H,
                      const float* __restrict__ W0, const float* __restrict__ bias0,
                      const float* __restrict__ W1, const float* __restrict__ bias1,
                      float* __restrict__ emb, float* __restrict__ ptrm) {
    const int l  = threadIdx.x;
    const int mt = blockIdx.x * 16;
    const int nt = blockIdx.y * 16;
    const float* W    = blockIdx.z ? W1    : W0;
    const float* bias = blockIdx.z ? bias1 : bias0;
    float*       out  = blockIdx.z ? ptrm  : emb;

    const int mr = l & 15;
    const int kh = (l >> 4) * 2;
    const float* arow = H + (size_t)(mt + mr) * EE + kh;
    const float* brow = W + (size_t)(nt + mr) * EE + kh;   // B[k][n] = W[n][k]

    v8f acc = {};
#pragma unroll 8
    for (int k0 = 0; k0 < EE; k0 += 4) {
        v2f a = *(const v2f*)(arow + k0);
        v2f b = *(const v2f*)(brow + k0);
        acc = __builtin_amdgcn_wmma_f32_16x16x4_f32(false, a, false, b,
                                                    (short)0, acc, false, false);
    }
    const float bv    = bias[nt + mr];
    const int   mbase = mt + 8 * (l >> 4);
#pragma unroll
    for (int j = 0; j < 8; ++j)
        out[(size_t)(mbase + j) * EE + (nt + mr)] = acc[j] + bv;
}

// ---------------------------------------------------------------------------
// Kernel 2: per batch b:
//   L1[s,n] = emb[b,s,:] . item_emb[seq[b,n],:]   (which==0)
//   P [s,k] = ptr[b,s,:] . H[b,k,:]               (which==1)
// M = N = 512, K = 128. One wave per 16x16 tile; B rows gathered by token.
// ---------------------------------------------------------------------------
__global__ __launch_bounds__(32)
void gemm_logits_wmma(const float* __restrict__ emb, const float* __restrict__ ptrm,
                      const float* __restrict__ hidden, const float* __restrict__ item_emb,
                      const int* __restrict__ seq,
                      float* __restrict__ L1, float* __restrict__ P) {
    const int l  = threadIdx.x;
    const int mt = blockIdx.x * 16;
    const int nt = blockIdx.y * 16;
    const int b     = blockIdx.z >> 1;
    const int which = blockIdx.z & 1;

    const int mr = l & 15;
    const int kh = (l >> 4) * 2;

    const float* A    = (which ? ptrm : emb) + (size_t)b * SS * EE;
    const float* arow = A + (size_t)(mt + mr) * EE + kh;

    const float* brow;
    if (which == 0) {
        const int tok = seq[b * SS + nt + mr];
        brow = item_emb + (size_t)tok * EE + kh;
    } else {
        brow = hidden + ((size_t)b * SS + (nt + mr)) * EE + kh;
    }

    v8f acc = {};
#pragma unroll 8
    for (int k0 = 0; k0 < EE; k0 += 4) {
        v2f a = *(const v2f*)(arow + k0);
        v2f bb = *(const v2f*)(brow + k0);
        acc = __builtin_amdgcn_wmma_f32_16x16x4_f32(false, a, false, bb,
                                                    (short)0, acc, false, false);
    }
    float* out = (which ? P : L1) + (size_t)b * SS * SS;
    const int mbase = mt + 8 * (l >> 4);
#pragma unroll
    for (int j = 0; j < 8; ++j)
        out[(size_t)(mbase + j) * SS + (nt + mr)] = acc[j];
}

// ---------------------------------------------------------------------------
// Kernel 3: one block per (b,s) row.
//   counts c = #{k<=s : seq[k]==seq[n]}
//   lh      = (sum_{k<=s, seq[k]==seq[n]} P[s,k]) / c   (0 if c==0)
//   val     = (L1[s,n] + lh) * log2(c+1)
//   out[b,s,seq[n]] = val   (duplicate tokens write the identical value,
//                            which equals the reference's scatter-mean)
// ---------------------------------------------------------------------------
__global__ __launch_bounds__(SS)
void row_scatter_kernel(const float* __restrict__ L1, const float* __restrict__ P,
                        const int* __restrict__ seq, float* __restrict__ out) {
    __shared__ int   s_seq[SS];
    __shared__ float s_p[SS];
    const int bs = blockIdx.x;          // b*S + s
    const int b  = bs / SS;
    const int s  = bs % SS;
    const int n  = threadIdx.x;

    s_seq[n] = seq[b * SS + n];
    s_p[n]   = P[(size_t)bs * SS + n];
    __syncthreads();

    const int tok = s_seq[n];
    int   c   = 0;
    float acc = 0.f;
    for (int k = 0; k <= s; ++k) {
        const bool m = (s_seq[k] == tok);
        c   += m ? 1 : 0;
        acc += m ? s_p[k] : 0.f;
    }
    const float lh  = (c > 0) ? acc / (float)c : 0.f;
    const float val = (L1[(size_t)bs * SS + n] + lh) * log2f((float)(c + 1));
    out[((size_t)b * SS + s) * VV + tok] = val;
}

// ---------------------------------------------------------------------------
// Host launcher.
// Inputs: 0 input_seq[int, B*S] 1 hidden[f32, B*S*E] 2 item_emb[f32, V*E]
//         3 W_emb[E*E] 4 b_emb[E] 5 W_ptr[E*E] 6 b_ptr[E]
// Output: f32 [B,S,V].
// Workspace layout (floats): emb | ptr | L1 | P  = 5.24 MB total.
// ---------------------------------------------------------------------------
extern "C" void kernel_launch(void* const* d_in, const int* in_sizes, int n_in,
                              void* d_out, int out_size, void* d_ws, size_t ws_size,
                              hipStream_t stream) {
    const int*   seq      = (const int*)  d_in[0];
    const float* hidden   = (const float*)d_in[1];
    const float* item_emb = (const float*)d_in[2];
    const float* W_emb    = (const float*)d_in[3];
    const float* b_emb    = (const float*)d_in[4];
    const float* W_ptr    = (const float*)d_in[5];
    const float* b_ptr    = (const float*)d_in[6];
    float*       out      = (float*)d_out;

    float* ws  = (float*)d_ws;
    float* emb = ws;                                   // B*S*E = 131072
    float* ptm = emb + (size_t)BB * SS * EE;           // 131072
    float* L1  = ptm + (size_t)BB * SS * EE;           // B*S*S = 524288
    float* P   = L1  + (size_t)BB * SS * SS;           // 524288

    // 1) stream zeros into the 204.8MB output (the bandwidth floor)
    const size_t n4 = (size_t)BB * SS * VV / 4;
    zero_out_kernel<<<4096, 256, 0, stream>>>(out, n4);

    // 2) emb / ptr linears: grid (M/16, N/16, 2), one wave per tile
    gemm_linear_wmma<<<dim3(BB * SS / 16, EE / 16, 2), 32, 0, stream>>>(
        hidden, W_emb, b_emb, W_ptr, b_ptr, emb, ptm);

    // 3) L1 and P: grid (S/16, S/16, 2*B)
    gemm_logits_wmma<<<dim3(SS / 16, SS / 16, 2 * BB), 32, 0, stream>>>(
        emb, ptm, hidden, item_emb, seq, L1, P);

    // 4) token-history + transform + scatter
    row_scatter_kernel<<<BB * SS, SS, 0, stream>>>(L1, P, seq, out);
}